// SinusoidalPositionalEncoding_87076166959279
// MI455X (gfx1250) — compile-verified
//
#include <hip/hip_runtime.h>
#include <math.h>

// Sinusoidal positional encoding add, MI455X (gfx1250, wave32).
// out[b,s,d] = x[b,s,d] + pe[s,d];  pe[s,2i]=sin(s*omega[i]), pe[s,2i+1]=cos(s*omega[i])
//
// Bandwidth-bound: 268 MB total traffic -> ~11.5 us at 23.3 TB/s HBM.
// One thread owns one float4 (= 2 sin/cos pairs) of one sequence position and
// streams all B batch images, so each sincos pair is computed once and reused
// 8x from registers. All global traffic is b128 non-temporal (single-touch
// streams; don't thrash the shared 192MB L2).
//
// Trig: angles reach 4095 rad, so instead of ocml's branchy precise sincosf
// (Payne-Hanek path is dynamically dead but costs ~100 VALU/SALU per pair),
// do a Cody-Waite reduction to fractional *revolutions* and feed the CDNA5
// hardware trans units (v_sin_f32 / v_cos_f32 take revolution units and
// co-execute with VALU). Reduction error ~1e-6 rad, below the f32 angle's
// own rounding error that the reference also carries.

#define D_MODEL 1024
#define SEQ_LEN 4096
#define D4      (D_MODEL / 4)      // 256 float4 per row
#define D4_LOG2 8

typedef float __attribute__((ext_vector_type(4))) v4f;
typedef float __attribute__((ext_vector_type(2))) v2f;

// 1/(2*pi) split into hi+lo floats at compile time (no device doubles at runtime).
constexpr double INV2PI_D = 0.15915494309189533576888376337251;
constexpr float  INV2PI_HI = (float)INV2PI_D;
constexpr float  INV2PI_LO = (float)(INV2PI_D - (double)INV2PI_HI);

// sin/cos of (ang radians) via fractional-revolution reduction + HW trans ops.
__device__ __forceinline__ void fast_sincos(float ang, float& s, float& c) {
    float rh = ang * INV2PI_HI;
    float rl = __builtin_fmaf(ang, INV2PI_HI, -rh);   // exact product residual
    rl       = __builtin_fmaf(ang, INV2PI_LO, rl);    // + low part of constant
    float n  = __builtin_rintf(rh);
    float fr = (rh - n) + rl;                         // [-0.5, 0.5] revolutions
    s = __builtin_amdgcn_sinf(fr);                    // v_sin_f32 (revolutions)
    c = __builtin_amdgcn_cosf(fr);                    // v_cos_f32 (revolutions)
}

__global__ __launch_bounds__(256) void sinpe_add_kernel(
    const v4f* __restrict__ x,       // [B*S*D/4]
    const v2f* __restrict__ omega2,  // [D/4] pairs of omega
    v4f* __restrict__ out,           // [B*S*D/4]
    int batch,
    long long rowStride4)            // S*D/4: float4 elems per batch image
{
    unsigned tid = blockIdx.x * blockDim.x + threadIdx.x;   // 0 .. S*D4-1
    unsigned f = tid & (D4 - 1);       // float4 column chunk: 0..255
    unsigned s = tid >> D4_LOG2;       // sequence position:   0..4095
    if (s >= SEQ_LEN) return;

    long long idx = (long long)s * D4 + f;

    // Start the first cacheline fetch before the trig preamble
    // (gfx1250 global_prefetch_b8; speculative, no counter cost).
    __builtin_prefetch(&x[idx], 0, 0);

    // omega pair for feature dims {4f..4f+3} -> pair indices {2f, 2f+1}
    v2f om = omega2[f];

    float p = (float)s;
    float s0, c0, s1, c1;
    fast_sincos(p * om.x, s0, c0);     // angle computed in f32, as reference does
    fast_sincos(p * om.y, s1, c1);
    const v4f pe = {s0, c0, s1, c1};

    // Stream all batch images: 8 independent b128 NT loads/stores.
#pragma unroll
    for (int b = 0; b < 8; ++b) {
        if (b >= batch) break;
        v4f v = __builtin_nontemporal_load(&x[idx]);
        v += pe;
        __builtin_nontemporal_store(v, &out[idx]);
        idx += rowStride4;
    }
}

extern "C" void kernel_launch(void* const* d_in, const int* in_sizes, int n_in,
                              void* d_out, int out_size, void* d_ws, size_t ws_size,
                              hipStream_t stream)
{
    const float* x     = (const float*)d_in[0];   // [8,4096,1024] f32
    const float* omega = (const float*)d_in[1];   // [512] f32
    float*       out   = (float*)d_out;

    const int D = 2 * in_sizes[1];                           // 1024
    const long long rows = (long long)in_sizes[0] / D;       // B*S
    const int batch = (int)(rows / SEQ_LEN);                 // 8
    const long long rowStride4 = (long long)SEQ_LEN * (D / 4);

    const int threads = SEQ_LEN * (D / 4);                   // 1,048,576
    dim3 block(256);
    dim3 grid((threads + block.x - 1) / block.x);            // 4096 blocks, 8 waves each

    sinpe_add_kernel<<<grid, block, 0, stream>>>(
        (const v4f*)x, (const v2f*)omega, (v4f*)out, batch, rowStride4);
}